// _CombinedGNNModel_52432960750007
// MI455X (gfx1250) — compile-verified
//
#include <hip/hip_runtime.h>

// Problem sizes (fixed by the reference)
#define NN   20000
#define EE   160000
#define CC   8
#define HH   64
#define OPE  64
#define FND  140
#define FCF  18
#define FIN  222          // OPE + FND + FCF
#define ROWS (NN*CC)      // 160000 rows of H=64

typedef float v2f __attribute__((ext_vector_type(2)));
typedef float v8f __attribute__((ext_vector_type(8)));

#if defined(__has_builtin)
#if __has_builtin(__builtin_amdgcn_wmma_f32_16x16x4_f32)
#define HAVE_WMMA_F32 1
#endif
#endif

__device__ __forceinline__ float leakyf(float v) { return v > 0.f ? v : 0.2f * v; }

// ---------------------------------------------------------------------------
// Fuse the 3 stacked linear layers (no nonlinearity between them — exact):
// Wf = W0@W1@W2 (222x64), bf = (b0@W1 + b1)@W2 + b2
// ---------------------------------------------------------------------------
__global__ void k_fuse(const float* __restrict__ W0, const float* __restrict__ b0,
                       const float* __restrict__ W1, const float* __restrict__ b1,
                       const float* __restrict__ W2, const float* __restrict__ b2,
                       float* __restrict__ Wf, float* __restrict__ bf) {
  __shared__ float T1[FIN * HH];   // 56832 B
  __shared__ float t1b[HH];
  const int h = threadIdx.x;       // 0..63
  for (int i = 0; i < FIN; ++i) {
    float s = 0.f;
    for (int k = 0; k < HH; ++k) s += W0[i * HH + k] * W1[k * HH + h];
    T1[i * HH + h] = s;
  }
  {
    float s = 0.f;
    for (int k = 0; k < HH; ++k) s += b0[k] * W1[k * HH + h];
    t1b[h] = s + b1[h];
  }
  __syncthreads();
  for (int i = 0; i < FIN; ++i) {
    float s = 0.f;
    for (int k = 0; k < HH; ++k) s += T1[i * HH + k] * W2[k * HH + h];
    Wf[i * HH + h] = s;
  }
  {
    float s = 0.f;
    for (int k = 0; k < HH; ++k) s += t1b[k] * W2[k * HH + h];
    bf[h] = s + b2[h];
  }
}

// float4 zero fill (all our buffers have length % 4 == 0)
__global__ void k_zero4(float4* __restrict__ p, int n4) {
  int i = blockIdx.x * blockDim.x + threadIdx.x;
  if (i < n4) { float4 z; z.x = z.y = z.z = z.w = 0.f; p[i] = z; }
}

__global__ void k_deg(const int* __restrict__ src, float* __restrict__ scl) {
  int e = blockIdx.x * blockDim.x + threadIdx.x;
  if (e < EE) atomicAdd(&scl[src[e]], 1.0f);
}

__global__ void k_scale(float* __restrict__ scl) {
  int n = blockIdx.x * blockDim.x + threadIdx.x;
  if (n < NN) scl[n] = 1.0f / (scl[n] + 1.0f);
}

// Per-node part of the fused input projection (shared by all 8 configs):
// base[n][h] = bf[h] + emb[op_ids[n]]·Wf[0:64] + node_feats[n]·Wf[64:204]
__global__ void k_base(const int* __restrict__ op_ids, const float* __restrict__ node_feats,
                       const float* __restrict__ emb, const float* __restrict__ Wf,
                       const float* __restrict__ bf, float* __restrict__ base) {
  __shared__ float row[OPE + FND];
  const int n = blockIdx.x, h = threadIdx.x;
  const int op = op_ids[n];
  for (int i = h; i < OPE; i += 64) row[i] = emb[op * OPE + i];
  for (int i = h; i < FND; i += 64) row[OPE + i] = node_feats[n * FND + i];
  __syncthreads();
  float s = bf[h];
  for (int k = 0; k < OPE + FND; ++k) s += row[k] * Wf[k * HH + h];
  base[n * HH + h] = s;
}

// Per-(node,config): x[r][h] = base[n][h] + config[r]·Wf[204:222]
__global__ void k_initx(const float* __restrict__ config, const float* __restrict__ Wf,
                        const float* __restrict__ base, float* __restrict__ x) {
  __shared__ float cf[FCF];
  const int r = blockIdx.x, h = threadIdx.x;
  const int n = r >> 3;
  if (h < FCF) cf[h] = config[r * FCF + h];
  __syncthreads();
  float s = base[n * HH + h];
  for (int k = 0; k < FCF; ++k) s += cf[k] * Wf[(OPE + FND + k) * HH + h];
  x[(size_t)r * HH + h] = s;
}

// Both directed aggregations carry scale[src] -> one shared atomic buffer:
//   agg[dst] += scale[src]*leaky(x[src])   (forward message)
//   agg[src] += scale[src]*leaky(x[dst])   (reverse message)
__global__ void k_scatter(const int* __restrict__ src, const int* __restrict__ dst,
                          const float* __restrict__ x, const float* __restrict__ scl,
                          float* __restrict__ agg) {
  const int e = blockIdx.x;      // one edge per 512-thread block (C*H = 512)
  const int o = threadIdx.x;
  const int s = src[e], d = dst[e];
  const float sc = scl[s];
  const float ys = leakyf(x[(size_t)s * 512 + o]) * sc;
  const float yd = leakyf(x[(size_t)d * 512 + o]) * sc;
  atomicAdd(&agg[(size_t)d * 512 + o], ys);
  atomicAdd(&agg[(size_t)s * 512 + o], yd);
}

// h[n] = agg[n] + leaky(x[n]) * (1 + 2*scale[n])   (in place, float4)
__global__ void k_assemble(const float4* __restrict__ x, const float* __restrict__ scl,
                           float4* __restrict__ agg) {
  const int i = blockIdx.x * blockDim.x + threadIdx.x;   // < ROWS*HH/4 = 2.56M
  const int n = i >> 7;                                  // 128 float4 per node-row block
  const float f = 1.f + 2.f * scl[n];
  float4 xv = x[i], av = agg[i];
  av.x += leakyf(xv.x) * f;
  av.y += leakyf(xv.y) * f;
  av.z += leakyf(xv.z) * f;
  av.w += leakyf(xv.w) * f;
  agg[i] = av;
}

// ---------------------------------------------------------------------------
// Fused shared-MLP: x += leaky(h @ sWa + sba) @ sWb + sbb, via f32 WMMA.
// 4 waves/block; each wave owns a 32-row slab (two 16-row M-tiles sharing every
// B-fragment). Weights staged in LDS once per block so B-fragments come from
// LDS, not global. 2 GEMMs x 16 K-steps x 4 N-tiles x 2 M-tiles = 256 WMMA/wave.
// Fragment layouts per ISA §7.12.2:
//   A 16x4 : elem(M=lane&15, K=v + 2*(lane>>4))
//   C/D    : elem(M=v + 8*(lane>>4), N=lane&15)
//   B 4x16 : assumed symmetric to A: elem(K=v + 2*(lane>>4), N=lane&15)
// ---------------------------------------------------------------------------
__global__ void __launch_bounds__(128) k_mlp(const float* __restrict__ hb,
                                             const float* __restrict__ Wa,
                                             const float* __restrict__ ba,
                                             const float* __restrict__ Wb,
                                             const float* __restrict__ bb,
                                             float* __restrict__ x) {
  __shared__ float WL[2][HH * HH];    // 32 KB: sWa, sWb
  __shared__ float tile[4][32 * HH];  // 32 KB: per-wave 32x64 h-tile, then t-tile
  const int tid = threadIdx.x;
  const int lane = tid & 31;
  const int wave = tid >> 5;
  const int r0 = blockIdx.x * 128 + wave * 32;

  // Stage both weight matrices to LDS (2048 float4 across 128 threads)
  {
    const float4* wa4 = (const float4*)Wa;
    const float4* wb4 = (const float4*)Wb;
    float4* l0 = (float4*)WL[0];
    float4* l1 = (float4*)WL[1];
#pragma unroll
    for (int i = 0; i < 8; ++i) {
      l0[i * 128 + tid] = wa4[i * 128 + tid];
      l1[i * 128 + tid] = wb4[i * 128 + tid];
    }
  }
  // Stage per-wave 32x64 h-tile (512 float4 across 32 lanes)
  float* T = tile[wave];
  {
    const float4* s4 = (const float4*)(hb + (size_t)r0 * HH);
    float4* t4 = (float4*)T;
#pragma unroll
    for (int i = 0; i < 16; ++i) t4[i * 32 + lane] = s4[i * 32 + lane];
  }
  __syncthreads();

  const int nl = lane & 15;
  const int kg = lane >> 4;
#if defined(HAVE_WMMA_F32)
  v8f acc0[4], acc1[4];
#pragma unroll
  for (int j = 0; j < 4; ++j) {
    const float b = ba[j * 16 + nl];
#pragma unroll
    for (int v = 0; v < 8; ++v) { acc0[j][v] = b; acc1[j][v] = b; }
  }
  // GEMM1: t = h @ sWa + sba
#pragma unroll
  for (int kk = 0; kk < 16; ++kk) {
    const int k0 = kk * 4 + 2 * kg;
    v2f a0; a0[0] = T[nl * HH + k0];        a0[1] = T[nl * HH + k0 + 1];
    v2f a1; a1[0] = T[(16 + nl) * HH + k0]; a1[1] = T[(16 + nl) * HH + k0 + 1];
#pragma unroll
    for (int j = 0; j < 4; ++j) {
      v2f b; b[0] = WL[0][k0 * HH + j * 16 + nl];
             b[1] = WL[0][(k0 + 1) * HH + j * 16 + nl];
      acc0[j] = __builtin_amdgcn_wmma_f32_16x16x4_f32(false, a0, false, b,
                                                      (short)0, acc0[j], false, false);
      acc1[j] = __builtin_amdgcn_wmma_f32_16x16x4_f32(false, a1, false, b,
                                                      (short)0, acc1[j], false, false);
    }
  }
  __syncthreads();
  // leaky + write t-tiles back to LDS (row-major 32x64)
#pragma unroll
  for (int j = 0; j < 4; ++j) {
#pragma unroll
    for (int v = 0; v < 8; ++v) {
      float t0 = acc0[j][v]; t0 = t0 > 0.f ? t0 : 0.2f * t0;
      float t1 = acc1[j][v]; t1 = t1 > 0.f ? t1 : 0.2f * t1;
      T[(v + 8 * kg) * HH + j * 16 + nl]        = t0;
      T[(16 + v + 8 * kg) * HH + j * 16 + nl]   = t1;
    }
  }
  __syncthreads();
  // GEMM2: m = t @ sWb + sbb  (reuse accumulators)
#pragma unroll
  for (int j = 0; j < 4; ++j) {
    const float b = bb[j * 16 + nl];
#pragma unroll
    for (int v = 0; v < 8; ++v) { acc0[j][v] = b; acc1[j][v] = b; }
  }
#pragma unroll
  for (int kk = 0; kk < 16; ++kk) {
    const int k0 = kk * 4 + 2 * kg;
    v2f a0; a0[0] = T[nl * HH + k0];        a0[1] = T[nl * HH + k0 + 1];
    v2f a1; a1[0] = T[(16 + nl) * HH + k0]; a1[1] = T[(16 + nl) * HH + k0 + 1];
#pragma unroll
    for (int j = 0; j < 4; ++j) {
      v2f b; b[0] = WL[1][k0 * HH + j * 16 + nl];
             b[1] = WL[1][(k0 + 1) * HH + j * 16 + nl];
      acc0[j] = __builtin_amdgcn_wmma_f32_16x16x4_f32(false, a0, false, b,
                                                      (short)0, acc0[j], false, false);
      acc1[j] = __builtin_amdgcn_wmma_f32_16x16x4_f32(false, a1, false, b,
                                                      (short)0, acc1[j], false, false);
    }
  }
  // x += m  (coalesced: lanes 0..15 cover one row's 16 consecutive floats)
#pragma unroll
  for (int j = 0; j < 4; ++j) {
#pragma unroll
    for (int v = 0; v < 8; ++v) {
      const size_t i0 = (size_t)(r0 + v + 8 * kg) * HH + j * 16 + nl;
      const size_t i1 = (size_t)(r0 + 16 + v + 8 * kg) * HH + j * 16 + nl;
      x[i0] += acc0[j][v];
      x[i1] += acc1[j][v];
    }
  }
#else
  // VALU fallback (keeps the file compiling if the f32 WMMA builtin is absent)
  float tv[64];
  for (int o = 0; o < 64; ++o) {
    const int idx = o * 32 + lane, M = idx >> 6, Nc = idx & 63;
    float s = ba[Nc];
    for (int k = 0; k < HH; ++k) s += T[M * HH + k] * WL[0][k * HH + Nc];
    tv[o] = s > 0.f ? s : 0.2f * s;
  }
  __syncthreads();
  for (int o = 0; o < 64; ++o) T[o * 32 + lane] = tv[o];
  __syncthreads();
  for (int o = 0; o < 64; ++o) {
    const int idx = o * 32 + lane, M = idx >> 6, Nc = idx & 63;
    float s = bb[Nc];
    for (int k = 0; k < HH; ++k) s += T[M * HH + k] * WL[1][k * HH + Nc];
    x[(size_t)(r0 + M) * HH + Nc] += s;
  }
#endif
}

// pooled[c*64+h] = sum_n leaky(x[n][c][h]); float4 over the 512-wide row
__global__ void k_pool(const float4* __restrict__ x, float* __restrict__ pooled) {
  const int t = threadIdx.x;                 // 0..127 -> 4 consecutive (c,h)
  const int n0 = blockIdx.x * 100, n1 = n0 + 100;
  float4 s; s.x = s.y = s.z = s.w = 0.f;
  for (int n = n0; n < n1; ++n) {
    float4 v = x[(size_t)n * 128 + t];
    s.x += leakyf(v.x); s.y += leakyf(v.y); s.z += leakyf(v.z); s.w += leakyf(v.w);
  }
  atomicAdd(&pooled[4 * t + 0], s.x);
  atomicAdd(&pooled[4 * t + 1], s.y);
  atomicAdd(&pooled[4 * t + 2], s.z);
  atomicAdd(&pooled[4 * t + 3], s.w);
}

__global__ void k_final(const float* __restrict__ pooled,
                        const float* __restrict__ pW1, const float* __restrict__ pb1,
                        const float* __restrict__ pW2, const float* __restrict__ pb2,
                        const float* __restrict__ pW3, const float* __restrict__ pb3,
                        float* __restrict__ out) {
  __shared__ float P[CC * HH], T1[CC * HH], T2[CC * HH];
  const int h = threadIdx.x;  // 64
  for (int i = h; i < CC * HH; i += 64) P[i] = pooled[i];
  __syncthreads();
  for (int c = 0; c < CC; ++c) {
    float s = pb1[h];
    for (int k = 0; k < HH; ++k) s += P[c * HH + k] * pW1[k * HH + h];
    T1[c * HH + h] = leakyf(s);
  }
  __syncthreads();
  for (int c = 0; c < CC; ++c) {
    float s = pb2[h];
    for (int k = 0; k < HH; ++k) s += T1[c * HH + k] * pW2[k * HH + h];
    T2[c * HH + h] = leakyf(s);
  }
  __syncthreads();
  if (h < CC) {
    float s = pb3[0];
    for (int k = 0; k < HH; ++k) s += T2[h * HH + k] * pW3[k];
    out[h] = s;
  }
}

extern "C" void kernel_launch(void* const* d_in, const int* in_sizes, int n_in,
                              void* d_out, int out_size, void* d_ws, size_t ws_size,
                              hipStream_t stream) {
  (void)in_sizes; (void)n_in; (void)out_size; (void)ws_size;
  const int*   op_ids = (const int*)d_in[0];
  const int*   src    = (const int*)d_in[1];
  const int*   dst    = (const int*)d_in[2];
  const float* node_f = (const float*)d_in[3];
  const float* config = (const float*)d_in[4];
  const float* emb    = (const float*)d_in[5];
  const float* W0 = (const float*)d_in[6];  const float* b0 = (const float*)d_in[7];
  const float* W1 = (const float*)d_in[8];  const float* b1 = (const float*)d_in[9];
  const float* W2 = (const float*)d_in[10]; const float* b2 = (const float*)d_in[11];
  const float* sWa = (const float*)d_in[12]; const float* sba = (const float*)d_in[13];
  const float* sWb = (const float*)d_in[14]; const float* sbb = (const float*)d_in[15];
  const float* pW1 = (const float*)d_in[16]; const float* pb1 = (const float*)d_in[17];
  const float* pW2 = (const float*)d_in[18]; const float* pb2 = (const float*)d_in[19];
  const float* pW3 = (const float*)d_in[20]; const float* pb3 = (const float*)d_in[21];
  float* out = (float*)d_out;

  // Workspace layout (floats): total 21,795,456 floats = 87.2 MB
  float* ws     = (float*)d_ws;
  float* Wf     = ws;                 // 14336 (222x64 used)
  float* bf     = ws + 14336;         // 128
  float* scl    = ws + 14464;         // 20480 (N used)
  float* base   = ws + 34944;         // 1,280,000 (N x 64)
  float* x      = ws + 1314944;       // 10,240,000 (ROWS x 64)
  float* agg    = ws + 11554944;      // 10,240,000
  float* pooled = ws + 21794944;      // 512

  // Fused input projection weights + degree scales
  k_fuse<<<1, 64, 0, stream>>>(W0, b0, W1, b1, W2, b2, Wf, bf);
  k_zero4<<<(NN / 4 + 255) / 256, 256, 0, stream>>>((float4*)scl, NN / 4);
  k_deg<<<(EE + 255) / 256, 256, 0, stream>>>(src, scl);
  k_scale<<<(NN + 255) / 256, 256, 0, stream>>>(scl);

  // x0 = concat(op_emb, node_feats, config_feats) @ Wf + bf
  k_base<<<NN, 64, 0, stream>>>(op_ids, node_f, emb, Wf, bf, base);
  k_initx<<<ROWS, 64, 0, stream>>>(config, Wf, base, x);

  // 3 GNN rounds
  for (int it = 0; it < 3; ++it) {
    k_zero4<<<(ROWS * HH / 4 + 255) / 256, 256, 0, stream>>>((float4*)agg, ROWS * HH / 4);
    k_scatter<<<EE, 512, 0, stream>>>(src, dst, x, scl, agg);
    k_assemble<<<(ROWS * HH / 4 + 255) / 256, 256, 0, stream>>>(
        (const float4*)x, scl, (float4*)agg);
    k_mlp<<<ROWS / 128, 128, 0, stream>>>(agg, sWa, sba, sWb, sbb, x);
  }

  // Pool + final MLP
  k_zero4<<<1, 128, 0, stream>>>((float4*)pooled, 128);
  k_pool<<<200, 128, 0, stream>>>((const float4*)x, pooled);
  k_final<<<1, 64, 0, stream>>>(pooled, pW1, pb1, pW2, pb2, pW3, pb3, out);
}